// HOI_56221121905229
// MI455X (gfx1250) — compile-verified
//
#include <hip/hip_runtime.h>

typedef float v2f __attribute__((ext_vector_type(2)));
typedef float v8f __attribute__((ext_vector_type(8)));

// Problem constants
#define K_CB   2048
#define C_DIM  64
#define HW     3136
#define NB_HW  196               // HW / 16
#define KP     (K_CB + 1)        // padded LDS row stride (2049) -> bank = (m+k)%64

// Dynamic LDS layout (bytes)
#define OFF_DIST  0
#define SZ_DIST   (16 * KP * 4)                 // 131136: prob/dist tile [16][2049]
#define OFF_AS    (OFF_DIST + SZ_DIST)          // A tile As[c][m], 64x16 f32
#define OFF_CN    (OFF_AS + 64 * 16 * 4)        // codebook norms [2048]
#define OFF_XN    (OFF_CN + K_CB * 4)           // pixel norms [16] (16B aligned)
#define OFF_PMIN  (OFF_XN + 64)                 // [seg][m] partial mins
#define OFF_PIDX  (OFF_PMIN + 16 * 16 * 4)
#define OFF_PSUM  (OFF_PIDX + 16 * 16 * 4)
#define OFF_RMIN  (OFF_PSUM + 16 * 16 * 4)
#define OFF_RINV  (OFF_RMIN + 64)
#define OFF_RIDX  (OFF_RINV + 64)
#define OFF_QP    (OFF_RIDX + 64)               // GEMM2 wave partials [4][32][8]
#define SMEM_BYTES (OFF_QP + 4 * 32 * 8 * 4)    // ~150.8 KB

// ---------------------------------------------------------------------------
// Prep: codebook squared norms -> ws; zero histogram output.
// ---------------------------------------------------------------------------
__global__ void vq_prep(const float* __restrict__ cb,
                        float* __restrict__ cnorm,
                        float* __restrict__ vq0) {
    int g = blockIdx.x * blockDim.x + threadIdx.x;
    if (g < K_CB) {
        const float4* row = (const float4*)(cb + (size_t)g * C_DIM);
        float s = 0.f;
        #pragma unroll
        for (int i = 0; i < C_DIM / 4; ++i) {
            float4 v = row[i];
            s += v.x * v.x + v.y * v.y + v.z * v.z + v.w * v.w;
        }
        cnorm[g] = s;
        vq0[g] = 0.f;
    }
}

// ---------------------------------------------------------------------------
// Main fused kernel: one block = 16 pixels x all 2048 codes.
// ---------------------------------------------------------------------------
__global__ __launch_bounds__(256) void vq_main(
    const float* __restrict__ feat, const float* __restrict__ cb,
    const float* __restrict__ cnorm,
    float* __restrict__ out_feat, float* __restrict__ assignment,
    float* __restrict__ distance, float* __restrict__ vq0) {

    extern __shared__ char smem[];
    float* distL = (float*)(smem + OFF_DIST);   // [16][KP]  dist -> prob (in place)
    float* As    = (float*)(smem + OFF_AS);     // [64][16]  As[c][m]
    float* cnL   = (float*)(smem + OFF_CN);     // [2048]
    float* xn    = (float*)(smem + OFF_XN);     // [16]
    float* pmin  = (float*)(smem + OFF_PMIN);   // [16][16]
    int*   pidx  = (int*)  (smem + OFF_PIDX);   // [16][16]
    float* psum  = (float*)(smem + OFF_PSUM);   // [16][16]
    float* rmin  = (float*)(smem + OFF_RMIN);   // [16]
    float* rinv  = (float*)(smem + OFF_RINV);   // [16]
    int*   ridx  = (int*)  (smem + OFF_RIDX);   // [16]
    float* qp    = (float*)(smem + OFF_QP);     // [4][32][8]

    const int tid  = threadIdx.x;
    const int lane = tid & 31;
    const int wv   = tid >> 5;                  // 8 waves
    const int n0   = blockIdx.x * 16;           // first pixel of tile
    const int b    = blockIdx.x / NB_HW;        // batch (tiles never straddle)
    const int hw0  = (blockIdx.x % NB_HW) * 16;

    // Warm codebook into L2 (gfx1250 global_prefetch_b8)
    for (int r = tid; r < K_CB; r += 256)
        __builtin_prefetch(cb + (size_t)r * C_DIM, 0, 3);

    // Stage codebook norms
    for (int k = tid; k < K_CB; k += 256) cnL[k] = cnorm[k];

    // Load A-tile: As[c][m] = feat[b,c,hw0+m]; 16B-aligned float4 both sides
    {
        int c = tid >> 2;                 // 0..63
        int m = (tid & 3) * 4;            // 0,4,8,12
        float4 v = *(const float4*)(feat + ((size_t)b * C_DIM + c) * HW + hw0 + m);
        *(float4*)(As + c * 16 + m) = v;
    }
    __syncthreads();

    if (tid < 16) {                       // per-pixel squared norm
        float s = 0.f;
        for (int c = 0; c < C_DIM; ++c) { float v = As[c * 16 + tid]; s += v * v; }
        xn[tid] = s;
    }
    __syncthreads();

    // ---- GEMM1: dot(16x2048) = A(16x64) x C^T(64x2048), fp32 WMMA -------
    const int mrow = lane & 15;           // M (A) or N (B/C/D) index
    const int khi  = (lane >> 4) * 2;     // K sub-offset: 0 (lanes 0-15) / 2 (16-31)

    // Loop-invariant per-lane row norms: lanes 0-15 -> xn[0..7], 16-31 -> xn[8..15]
    // (two ds_load_b128, kills the per-element LDS round-trip in the epilogue)
    float xnv[8];
    {
        float4 x0 = *(const float4*)(xn + 8 * (lane >> 4));
        float4 x1 = *(const float4*)(xn + 8 * (lane >> 4) + 4);
        xnv[0] = x0.x; xnv[1] = x0.y; xnv[2] = x0.z; xnv[3] = x0.w;
        xnv[4] = x1.x; xnv[5] = x1.y; xnv[6] = x1.z; xnv[7] = x1.w;
    }

    for (int kt = 0; kt < 16; ++kt) {
        const int kb = wv * 256 + kt * 16;            // k-tile base
        v8f acc = {};
        const float* brow = cb + (size_t)(kb + mrow) * C_DIM + khi; // B: C^T[k, n]=C[n, k]
        #pragma unroll
        for (int c0 = 0; c0 < C_DIM; c0 += 4) {
            v2f a, bb;
            a.x = As[(c0 + khi) * 16 + mrow];          // A[M, c0+khi]
            a.y = As[(c0 + khi + 1) * 16 + mrow];      // A[M, c0+khi+1]
            bb  = *(const v2f*)(brow + c0);            // 8B-aligned (khi even)
            acc = __builtin_amdgcn_wmma_f32_16x16x4_f32(
                      false, a, false, bb, (short)0, acc, false, false);
        }
        // distance = ||x||^2 + ||c||^2 - 2*dot -> LDS (register-only math now)
        const int   kcol = kb + mrow;
        const float cnk  = cnL[kcol];
        #pragma unroll
        for (int r = 0; r < 8; ++r) {
            const int M = r + 8 * (lane >> 4);
            distL[M * KP + kcol] = (xnv[r] + cnk) - 2.f * acc[r];
        }
    }
    __syncthreads();

    // ---- Softmax stats: row min (+ first-occurrence argmin), exp-sum ----
    {
        const int m = tid & 15, seg = tid >> 4;
        float mn = 3.4e38f; int mi = 0;
        for (int j = 0; j < 128; ++j) {
            int k = seg * 128 + j;
            float d = distL[m * KP + k];
            if (d < mn) { mn = d; mi = k; }
        }
        pmin[seg * 16 + m] = mn; pidx[seg * 16 + m] = mi;
    }
    __syncthreads();
    if (tid < 16) {
        float mn = 3.4e38f; int mi = 0;
        for (int s = 0; s < 16; ++s) {
            float v = pmin[s * 16 + tid];
            if (v < mn) { mn = v; mi = pidx[s * 16 + tid]; }
        }
        rmin[tid] = mn; ridx[tid] = mi;
    }
    __syncthreads();
    {
        const int m = tid & 15, seg = tid >> 4;
        const float rm = rmin[m];
        float s = 0.f;
        for (int j = 0; j < 128; ++j)
            s += __expf((rm - distL[m * KP + seg * 128 + j]) * 10.f);
        psum[seg * 16 + m] = s;
    }
    __syncthreads();
    if (tid < 16) {
        float s = 0.f;
        for (int sg = 0; sg < 16; ++sg) s += psum[sg * 16 + tid];
        rinv[tid] = 1.f / s;
        atomicAdd(vq0 + ridx[tid], 1.0f);   // integer-valued f32 add: exact, order-safe
    }
    __syncthreads();

    // ---- Write distance (coalesced), convert LDS tile to prob in place ----
    {
        const int m = tid >> 4, li = tid & 15;     // 16 lanes -> 64B runs
        const float rm = rmin[m], ri = rinv[m];
        float* drow = distance + (size_t)(n0 + m) * K_CB;
        for (int j = 0; j < 128; ++j) {
            int k = li + 16 * j;
            float d = distL[m * KP + k];
            drow[k] = d;
            distL[m * KP + k] = __expf((rm - d) * 10.f) * ri;
        }
    }
    __syncthreads();

    // ---- GEMM2: q(16x64) = prob(16x2048) x C(2048x64), fp32 WMMA --------
    {
        const int ct = wv & 3;                 // c-tile 0..3 (16 channels)
        const int kh = wv >> 2;                // K half: 0 / 1
        const int ccol = ct * 16 + mrow;       // output channel (B/D column)
        v8f acc = {};
        for (int s = 0; s < 256; ++s) {
            const int kk = kh * 1024 + s * 4 + khi;
            v2f a, bb;
            a.x  = distL[mrow * KP + kk];                    // prob[M, kk]
            a.y  = distL[mrow * KP + kk + 1];
            bb.x = cb[(size_t)kk * C_DIM + ccol];            // C[kk, ccol]
            bb.y = cb[(size_t)(kk + 1) * C_DIM + ccol];
            acc = __builtin_amdgcn_wmma_f32_16x16x4_f32(
                      false, a, false, bb, (short)0, acc, false, false);
        }
        if (wv >= 4) {                        // stash upper-half partials
            float* q = qp + ((wv - 4) * 32 + lane) * 8;
            #pragma unroll
            for (int r = 0; r < 8; ++r) q[r] = acc[r];
        }
        __syncthreads();
        if (wv < 4) {                         // reduce + scatter out_feat (6.4MB total)
            const float* q = qp + (wv * 32 + lane) * 8;
            float* obase = out_feat + ((size_t)b * C_DIM + ccol) * HW + hw0;
            #pragma unroll
            for (int r = 0; r < 8; ++r) {
                const int M = r + 8 * (lane >> 4);
                obase[M] = acc[r] + q[r];
            }
        }
    }

    // ---- assignment[b,k,hw0+m] = prob[m,k] (16 lanes = contiguous 64B) ----
    // After GEMM2: waves 4-7 clear the reduction barrier first and stream these
    // stores while waves 0-3 finish out_feat.
    {
        const int m = tid & 15;
        float* abase = assignment + (size_t)b * K_CB * HW + hw0 + m;
        for (int j = 0; j < 128; ++j) {
            int k = (tid >> 4) + 16 * j;
            abase[(size_t)k * HW] = distL[m * KP + k];
        }
    }
}

// ---------------------------------------------------------------------------
extern "C" void kernel_launch(void* const* d_in, const int* in_sizes, int n_in,
                              void* d_out, int out_size, void* d_ws, size_t ws_size,
                              hipStream_t stream) {
    const float* feat = (const float*)d_in[0];   // (8,64,56,56)
    const float* cb   = (const float*)d_in[1];   // (2048,64)
    // d_in[2] = cur_iter (unused: reference computes histogram unconditionally)

    float* out        = (float*)d_out;
    float* out_feat   = out;                                 // 1,605,632
    float* assignment = out_feat + (size_t)1605632;          // 51,380,224
    float* distance   = assignment + (size_t)51380224;       // 51,380,224
    float* vq0        = distance + (size_t)51380224;         // 2,048
    float* cnorm      = (float*)d_ws;                        // 2048 f32 scratch

    vq_prep<<<8, 256, 0, stream>>>(cb, cnorm, vq0);
    vq_main<<<25088 / 16, 256, SMEM_BYTES, stream>>>(
        feat, cb, cnorm, out_feat, assignment, distance, vq0);
}